// GAT2Layers_56624848830740
// MI455X (gfx1250) — compile-verified
//
#include <hip/hip_runtime.h>
#include <math.h>

typedef __attribute__((ext_vector_type(2))) float v2f;
typedef __attribute__((ext_vector_type(8))) float v8f;

#define FDIM 128
#define NHEAD 4

// ---- monotonic float<->u32 encoding so atomicMax(u32) == float max ----
__device__ __forceinline__ unsigned enc_f32(float f) {
    unsigned u = __float_as_uint(f);
    return (u & 0x80000000u) ? ~u : (u | 0x80000000u);
}
__device__ __forceinline__ float dec_f32(unsigned e) {
    unsigned u = (e & 0x80000000u) ? (e ^ 0x80000000u) : ~e;
    return __uint_as_float(u);
}

// ---------------------------------------------------------------------
// GEMM: out = x @ W + bias for both (Wl,bl)->xl and (Wr,br)->xr.
// One wave per 16-row tile; 8 f32 accumulators cover all 128 columns.
// V_WMMA_F32_16X16X4_F32, K stepped by 4, 32 steps over K=128.
// A layout: lanes 0-15 rows M=0..15 K={k,k+1}; lanes 16-31 K={k+2,k+3}.
// B layout mirrors A with N striped across lanes.
// ---------------------------------------------------------------------
__global__ __launch_bounds__(256) void gemm_xlxr(
    const float* __restrict__ x,
    const float* __restrict__ Wl, const float* __restrict__ bl,
    const float* __restrict__ Wr, const float* __restrict__ br,
    float* __restrict__ xl, float* __restrict__ xr, int N)
{
    const int wave = (int)((blockIdx.x * blockDim.x + threadIdx.x) >> 5);
    const int lane = (int)(threadIdx.x & 31u);
    const int row0 = wave * 16;
    if (row0 >= N) return;

    const float* __restrict__ W    = blockIdx.y ? Wr : Wl;
    const float* __restrict__ bias = blockIdx.y ? br : bl;
    float* __restrict__ out        = blockIdx.y ? xr : xl;

    const int m15  = lane & 15;
    const int koff = (lane < 16) ? 0 : 2;
    const float* __restrict__ xrow = x + (size_t)(row0 + m15) * FDIM;

    v8f acc[8] = {};
    for (int k = 0; k < FDIM; k += 4) {
        v2f a;
        a.x = xrow[k + koff];
        a.y = xrow[k + koff + 1];
        const float* __restrict__ w0 = W + (size_t)(k + koff) * FDIM + m15;
#pragma unroll
        for (int t = 0; t < 8; ++t) {
            v2f b;
            b.x = w0[t * 16];
            b.y = w0[t * 16 + FDIM];
            acc[t] = __builtin_amdgcn_wmma_f32_16x16x4_f32(
                false, a, false, b, (short)0, acc[t], false, false);
        }
    }
    // D layout: VGPR r -> row (r | lanes>=16 ? +8), col = lane&15 within tile
    const int rbase = (lane < 16) ? 0 : 8;
#pragma unroll
    for (int t = 0; t < 8; ++t) {
        const int col = t * 16 + m15;
        const float bv = bias[col];
#pragma unroll
        for (int r = 0; r < 8; ++r)
            out[(size_t)(row0 + rbase + r) * FDIM + col] = acc[t][r] + bv;
    }
}

// ---------------------------------------------------------------------
// Init: out[n, :] = bias (final +b folded into the accumulator init),
// amax_enc = 0 (== -inf under encoding), denom = 0.
// ---------------------------------------------------------------------
__global__ __launch_bounds__(256) void init_layer(
    float* __restrict__ out, const float* __restrict__ bias,
    unsigned* __restrict__ amax, float* __restrict__ denom, int N)
{
    const int i = blockIdx.x * blockDim.x + threadIdx.x;
    if (i < N * FDIM) out[i] = bias[i & (FDIM - 1)];
    if (i < N * NHEAD) { amax[i] = 0u; denom[i] = 0.0f; }
}

// ---------------------------------------------------------------------
// Edge logits: wave per edge. Each lane handles 4 channels (float4),
// head = lane>>3 (8 lanes per head). a[e,h] = att_h . leaky(xl[src]+xr[dst]).
// Also atomic segment-max into amax_enc[dst,h].
// ---------------------------------------------------------------------
__global__ __launch_bounds__(256) void edge_logits(
    const float* __restrict__ xl, const float* __restrict__ xr,
    const float* __restrict__ att,
    const int* __restrict__ src, const int* __restrict__ dst,
    float* __restrict__ a, unsigned* __restrict__ amax, int E)
{
    const int e = (int)((blockIdx.x * blockDim.x + threadIdx.x) >> 5);
    if (e >= E) return;
    const int lane = (int)(threadIdx.x & 31u);
    const int s = src[e], d = dst[e];

    float4 m = ((const float4*)(xl + (size_t)s * FDIM))[lane];
    float4 r = ((const float4*)(xr + (size_t)d * FDIM))[lane];
    float4 v;
    v.x = m.x + r.x; v.x = v.x > 0.f ? v.x : 0.2f * v.x;
    v.y = m.y + r.y; v.y = v.y > 0.f ? v.y : 0.2f * v.y;
    v.z = m.z + r.z; v.z = v.z > 0.f ? v.z : 0.2f * v.z;
    v.w = m.w + r.w; v.w = v.w > 0.f ? v.w : 0.2f * v.w;
    float4 at = ((const float4*)att)[lane];  // flat idx 4*lane == h*32 + c
    float p = v.x * at.x + v.y * at.y + v.z * at.z + v.w * at.w;
    p += __shfl_xor(p, 1, 32);
    p += __shfl_xor(p, 2, 32);
    p += __shfl_xor(p, 4, 32);
    if ((lane & 7) == 0) {
        const int h = lane >> 3;
        a[(size_t)e * NHEAD + h] = p;
        atomicMax(&amax[(size_t)d * NHEAD + h], enc_f32(p));
    }
}

// ---------------------------------------------------------------------
// exp(a - amax[dst]) and atomic segment-sum into denom. Thread per (e,h).
// ---------------------------------------------------------------------
__global__ __launch_bounds__(256) void edge_exp(
    float* __restrict__ a, const unsigned* __restrict__ amax,
    const int* __restrict__ dst, float* __restrict__ denom, int E)
{
    const int i = blockIdx.x * blockDim.x + threadIdx.x;
    if (i >= E * NHEAD) return;
    const int e = i >> 2, h = i & 3;
    const int d = dst[e];
    const float mx = dec_f32(amax[(size_t)d * NHEAD + h]);
    const float ex = __expf(a[i] - mx);
    a[i] = ex;
    atomicAdd(&denom[(size_t)d * NHEAD + h], ex);
}

// ---------------------------------------------------------------------
// Aggregate: wave per edge. w = a[e,h]/(denom[dst,h]+1e-16);
// out[dst, c] += w * xl[src, c] via f32 atomics (resolved in L2).
// ---------------------------------------------------------------------
__global__ __launch_bounds__(256) void edge_aggregate(
    const float* __restrict__ xl, const float* __restrict__ a,
    const float* __restrict__ denom,
    const int* __restrict__ src, const int* __restrict__ dst,
    float* __restrict__ out, int E)
{
    const int e = (int)((blockIdx.x * blockDim.x + threadIdx.x) >> 5);
    if (e >= E) return;
    const int lane = (int)(threadIdx.x & 31u);
    const int s = src[e], d = dst[e];
    const int h = lane >> 3;
    const float w = a[(size_t)e * NHEAD + h] /
                    (denom[(size_t)d * NHEAD + h] + 1e-16f);
    float4 m = ((const float4*)(xl + (size_t)s * FDIM))[lane];
    float* o = out + (size_t)d * FDIM + lane * 4;
    atomicAdd(o + 0, w * m.x);
    atomicAdd(o + 1, w * m.y);
    atomicAdd(o + 2, w * m.z);
    atomicAdd(o + 3, w * m.w);
}

__global__ __launch_bounds__(256) void relu_inplace(float* __restrict__ p, int n)
{
    const int i = blockIdx.x * blockDim.x + threadIdx.x;
    if (i < n) p[i] = fmaxf(p[i], 0.0f);
}

// ---------------------------------------------------------------------
extern "C" void kernel_launch(void* const* d_in, const int* in_sizes, int n_in,
                              void* d_out, int out_size, void* d_ws, size_t ws_size,
                              hipStream_t stream) {
    // setup_inputs() dict order:
    const float* x    = (const float*)d_in[0];
    const int*   ei   = (const int*)  d_in[1];
    const float* Wl1  = (const float*)d_in[2];
    const float* Wr1  = (const float*)d_in[3];
    const float* Wl2  = (const float*)d_in[4];
    const float* Wr2  = (const float*)d_in[5];
    const float* bl1  = (const float*)d_in[6];
    const float* br1  = (const float*)d_in[7];
    const float* bl2  = (const float*)d_in[8];
    const float* br2  = (const float*)d_in[9];
    const float* att1 = (const float*)d_in[10];
    const float* att2 = (const float*)d_in[11];
    const float* b1   = (const float*)d_in[12];
    const float* b2   = (const float*)d_in[13];

    const int N = in_sizes[0] / FDIM;
    const int E = in_sizes[1] / 2;
    const int* src = ei;
    const int* dst = ei + E;

    // workspace layout (floats): xl | xr | h1 | a | amax | denom
    float*    ws    = (float*)d_ws;
    float*    xl    = ws;
    float*    xr    = xl + (size_t)N * FDIM;
    float*    h1    = xr + (size_t)N * FDIM;
    float*    a     = h1 + (size_t)N * FDIM;
    unsigned* amax  = (unsigned*)(a + (size_t)E * NHEAD);
    float*    denom = (float*)(amax + (size_t)N * NHEAD);

    const dim3 blk(256);
    const dim3 gemmGrid(((N + 15) / 16 + 7) / 8, 2);     // 8 waves/block, y: l/r
    const int  nfBlocks   = (N * FDIM + 255) / 256;
    const int  edgeWaves  = (E * 32 + 255) / 256;        // wave per edge
    const int  ehBlocks   = (E * NHEAD + 255) / 256;

    // ---------------- layer 1 ----------------
    gemm_xlxr<<<gemmGrid, blk, 0, stream>>>(x, Wl1, bl1, Wr1, br1, xl, xr, N);
    init_layer<<<nfBlocks, blk, 0, stream>>>(h1, b1, amax, denom, N);
    edge_logits<<<edgeWaves, blk, 0, stream>>>(xl, xr, att1, src, dst, a, amax, E);
    edge_exp<<<ehBlocks, blk, 0, stream>>>(a, amax, dst, denom, E);
    edge_aggregate<<<edgeWaves, blk, 0, stream>>>(xl, a, denom, src, dst, h1, E);
    relu_inplace<<<nfBlocks, blk, 0, stream>>>(h1, N * FDIM);

    // ---------------- layer 2 ----------------
    float* out = (float*)d_out;
    gemm_xlxr<<<gemmGrid, blk, 0, stream>>>(h1, Wl2, bl2, Wr2, br2, xl, xr, N);
    init_layer<<<nfBlocks, blk, 0, stream>>>(out, b2, amax, denom, N);
    edge_logits<<<edgeWaves, blk, 0, stream>>>(xl, xr, att2, src, dst, a, amax, E);
    edge_exp<<<ehBlocks, blk, 0, stream>>>(a, amax, dst, denom, E);
    edge_aggregate<<<edgeWaves, blk, 0, stream>>>(xl, a, denom, src, dst, out, E);
}